// GRUDecoder_51814485459463
// MI455X (gfx1250) — compile-verified
//
#include <hip/hip_runtime.h>
#include <hip/hip_bf16.h>

// GRU decoder for MI455X (gfx1250), wave32 + V_WMMA_F32_16X16X4_F32.
// Phase 1: x_proj = h_enc @ W_ih^T + b_ih   (parallel WMMA GEMM, 100.7 GFLOP)
// Phase 2: persistent 32-block scan, W_hh slice resident in LDS (96KB/WGP),
//          device-wide step sync via agent-scope atomics.

typedef float v2f __attribute__((ext_vector_type(2)));
typedef float v4f __attribute__((ext_vector_type(4)));
typedef float v8f __attribute__((ext_vector_type(8)));

#define B_    64
#define T_    1000
#define N_    512
#define G3_   1536          // 3*N
#define GBLK  32            // blocks in the scan kernel (must all be resident)
#define NCB   16            // hidden columns per scan block (512/32)
#define WROWS 48            // 3 gates * 16 columns of W_hh staged per block
#define WPITCH 516          // LDS row pitch (floats): 512+4 -> conflict-free b64

// ---------------------------------------------------------------------------
// Phase 0: zero the per-timestep arrival flags (must be re-zeroed every call)
// ---------------------------------------------------------------------------
__global__ void gru_init_kernel(unsigned int* __restrict__ flags) {
    int i = blockIdx.x * 256 + threadIdx.x;
    if (i < T_) flags[i] = 0u;
}

// ---------------------------------------------------------------------------
// Phase 1: x_proj[r][g] = sum_k h_enc[r][k] * W_ih[g][k] + b_ih[g]
//   r in [0, B*T), g in [0, 3N). 8 waves/block, each wave one 16x16 tile,
//   block covers 64 rows x 32 cols. K=512 via 128 x wmma_f32_16x16x4_f32.
// ---------------------------------------------------------------------------
__global__ void __launch_bounds__(256)
gru_xproj_kernel(const float* __restrict__ h_enc,
                 const float* __restrict__ W_ih,
                 const float* __restrict__ b_ih,
                 float* __restrict__ xp) {
    const int lane = threadIdx.x & 31;
    const int wave = threadIdx.x >> 5;
    const int mw = wave & 3;            // 4 M-tiles
    const int nw = wave >> 2;           // 2 N-tiles
    const int m0 = blockIdx.y * 64 + mw * 16;   // row-tile base in [0, 64000)
    const int g0 = blockIdx.x * 32 + nw * 16;   // col-tile base in [0, 1536)

    const int rn  = lane & 15;          // A row / B col / C col within tile
    const int khi = (lane >> 4) << 1;   // K sub-pair offset (0 or 2)

    // A: 16x4 tile of h_enc rows; B: 4x16 tile = W_ih rows (since W_ih^T)
    const float* Ap = h_enc + (size_t)(m0 + rn) * N_ + khi;
    const float* Bp = W_ih  + (size_t)(g0 + rn) * N_ + khi;

    v8f acc = {};
#pragma unroll 8
    for (int kb = 0; kb < N_; kb += 4) {
        v2f a = *(const v2f*)(Ap + kb);
        v2f b = *(const v2f*)(Bp + kb);
        acc = __builtin_amdgcn_wmma_f32_16x16x4_f32(
                  false, a, false, b, (short)0, acc, false, false);
    }

    const float bias = b_ih[g0 + rn];
    const int hi = lane >> 4;
#pragma unroll
    for (int v = 0; v < 8; ++v) {
        int r = m0 + v + 8 * hi;        // C/D layout: row = v + 8*(lane>>4)
        xp[(size_t)r * G3_ + g0 + rn] = acc[v] + bias;
    }
}

// ---------------------------------------------------------------------------
// Phase 2: persistent scan. 32 blocks x 128 threads (4 waves).
//   Block b owns hidden columns [b*16, b*16+16). Stages its 48 W_hh rows
//   (r,z,n gates) into LDS once; per step computes 3 gate tiles per wave
//   (wave = M-tile of 16 batch rows), keeps h in registers, ping-pongs the
//   full h through a compact global buffer, syncs all blocks via flags[t].
// ---------------------------------------------------------------------------
__global__ void __launch_bounds__(128)
gru_scan_kernel(const float* __restrict__ xp,
                const float* __restrict__ W_hh,
                const float* __restrict__ b_hh,
                float* __restrict__ out,      // [B][T][N]
                float* __restrict__ hbuf,     // [2][B][N] ping-pong
                unsigned int* __restrict__ flags) {
    extern __shared__ float Ws[];             // [WROWS][WPITCH]
    const int tid  = threadIdx.x;
    const int lane = tid & 31;
    const int wave = tid >> 5;                // 0..3 -> M tile
    const int nc0  = blockIdx.x * NCB;        // this block's hidden-col base

    // Stage W_hh slice: LDS row (j*16+c) = W_hh[j*512 + nc0 + c][:], j=gate
    for (int idx = tid; idx < WROWS * (N_ / 4); idx += 128) {
        int r  = idx >> 7;                    // N_/4 == 128 chunks per row
        int c4 = (idx & 127) * 4;
        int j  = r / NCB, c = r - j * NCB;
        v4f val = *(const v4f*)(W_hh + (size_t)(j * N_ + nc0 + c) * N_ + c4);
        *(v4f*)(Ws + r * WPITCH + c4) = val;
    }
    __syncthreads();

    const int n   = lane & 15;                // column within tile (and A row)
    const int khi = (lane >> 4) << 1;
    const int hi  = lane >> 4;
    const int m0  = wave * 16;                // batch-row tile base

    const float bhr = b_hh[          nc0 + n];
    const float bhz = b_hh[    N_  + nc0 + n];
    const float bhn = b_hh[2 * N_  + nc0 + n];

    const float* w0 = Ws + (0 * NCB + n) * WPITCH + khi;
    const float* w1 = Ws + (1 * NCB + n) * WPITCH + khi;
    const float* w2 = Ws + (2 * NCB + n) * WPITCH + khi;

    v8f hreg = {};                            // h for (my 8 rows, my column)

    for (int t = 0; t < T_; ++t) {
        v8f accR = {}, accZ = {}, accN = {};
        if (t > 0) {
            // wait until every block published h_{t-1}
            if (tid == 0) {
                while (__hip_atomic_load(&flags[t - 1], __ATOMIC_RELAXED,
                                         __HIP_MEMORY_SCOPE_AGENT) < GBLK)
                    __builtin_amdgcn_s_sleep(1);
            }
            __syncthreads();
            __threadfence();                  // acquire: invalidate L0s

            const float* hp = hbuf + (size_t)((t - 1) & 1) * (B_ * N_)
                                   + (size_t)(m0 + n) * N_ + khi;
#pragma unroll 8
            for (int kb = 0; kb < N_; kb += 4) {
                v2f a  = *(const v2f*)(hp + kb);     // global (L2-hot)
                v2f br = *(const v2f*)(w0 + kb);     // LDS
                v2f bz = *(const v2f*)(w1 + kb);
                v2f bn = *(const v2f*)(w2 + kb);
                accR = __builtin_amdgcn_wmma_f32_16x16x4_f32(
                           false, a, false, br, (short)0, accR, false, false);
                accZ = __builtin_amdgcn_wmma_f32_16x16x4_f32(
                           false, a, false, bz, (short)0, accZ, false, false);
                accN = __builtin_amdgcn_wmma_f32_16x16x4_f32(
                           false, a, false, bn, (short)0, accN, false, false);
            }
        }

        // Elementwise GRU update for my 8 (row, column) cells
        float* hb = hbuf + (size_t)(t & 1) * (B_ * N_);
#pragma unroll
        for (int v = 0; v < 8; ++v) {
            int m = m0 + v + 8 * hi;                 // batch row
            size_t xrow = ((size_t)m * T_ + t) * (size_t)G3_;
            float xr = xp[xrow +           nc0 + n];
            float xz = xp[xrow +     N_  + nc0 + n];
            float xn = xp[xrow + 2 * N_  + nc0 + n];
            float r  = 1.0f / (1.0f + __expf(-(xr + accR[v] + bhr)));
            float z  = 1.0f / (1.0f + __expf(-(xz + accZ[v] + bhz)));
            float g  = tanhf(xn + r * (accN[v] + bhn));
            float hn = (1.0f - z) * g + z * hreg[v];
            hreg[v] = hn;
            hb[(size_t)m * N_ + nc0 + n] = hn;
            out[((size_t)m * T_ + t) * (size_t)N_ + nc0 + n] = hn;
        }

        __threadfence();                      // release: flush to L2
        __syncthreads();
        if (tid == 0)
            __hip_atomic_fetch_add(&flags[t], 1u, __ATOMIC_RELAXED,
                                   __HIP_MEMORY_SCOPE_AGENT);
    }
}

// ---------------------------------------------------------------------------
extern "C" void kernel_launch(void* const* d_in, const int* in_sizes, int n_in,
                              void* d_out, int out_size, void* d_ws, size_t ws_size,
                              hipStream_t stream) {
    const float* h_enc = (const float*)d_in[0];   // [B,T,N]
    const float* W_ih  = (const float*)d_in[1];   // [3N,N]
    const float* W_hh  = (const float*)d_in[2];   // [3N,N]
    const float* b_ih  = (const float*)d_in[3];   // [3N]
    const float* b_hh  = (const float*)d_in[4];   // [3N]
    float* out = (float*)d_out;                   // [B,T,N]

    // Workspace layout: x_proj [B*T,3N] | hbuf [2,B,N] | flags [T]
    float* xp   = (float*)d_ws;
    float* hbuf = xp + (size_t)B_ * T_ * G3_;
    unsigned int* flags = (unsigned int*)(hbuf + 2 * B_ * N_);

    gru_init_kernel<<<(T_ + 255) / 256, 256, 0, stream>>>(flags);

    dim3 gA(G3_ / 32, (B_ * T_) / 64);
    gru_xproj_kernel<<<gA, 256, 0, stream>>>(h_enc, W_ih, b_ih, xp);

    const int smem = WROWS * WPITCH * (int)sizeof(float);   // 99072 B
    (void)hipFuncSetAttribute((const void*)gru_scan_kernel,
                              hipFuncAttributeMaxDynamicSharedMemorySize, smem);
    gru_scan_kernel<<<GBLK, 128, smem, stream>>>(xp, W_hh, b_hh, out, hbuf, flags);
}